// GaussianKernelConv_76158360092778
// MI455X (gfx1250) — compile-verified
//
#include <hip/hip_runtime.h>

typedef __attribute__((ext_vector_type(2))) float v2f;
typedef __attribute__((ext_vector_type(8))) float v8f;

namespace {
constexpr int kBN           = 8 * 8192;  // (B*N) pairs, one wave each
constexpr int kK            = 32;        // neighbors per point
constexpr int kP            = 16;        // kernel points
constexpr int kWavesPerBlk  = 8;
constexpr int kBlockDim     = 32 * kWavesPerBlk;
}

// out[b,n,p] = mean_k exp( -(||x_{b,n,k} - kp_p||^2) / (2 sigma^2) )
//
// One wave per (b,n). Two V_WMMA_F32_16X16X4_F32 per wave compute the full
// Gaussian exponent for 16 neighbor rows x 16 kernel points each:
//   A[m][0..2] = x_m,  A[m][3] = ||x_m||^2
//   B[k][p]    = 2s*kp[p][k] (k<3),  B[3][p] = -s      (s = 1/(2 sigma^2))
//   C[m][p]    = -s*||kp_p||^2
// => D[m][p]  = -s*(||x||^2 + ||kp||^2 - 2 x.kp)
__global__ __launch_bounds__(kBlockDim)
void gaussian_kernel_corr(const float* __restrict__ X,   // (B*N*K, 3) row-major
                          const float* __restrict__ KP,  // (P, 3)
                          const float* __restrict__ sigma_ptr,
                          float* __restrict__ out)       // (B*N, P)
{
  const int lane = threadIdx.x & 31;
  const int wave = blockIdx.x * kWavesPerBlk + (threadIdx.x >> 5);
  const int p    = lane & 15;
  const bool hi  = lane >= 16;

  const float sigma = sigma_ptr[0];
  const float s = 1.0f / (2.0f * sigma * sigma);

  // ---- B fragment: 4x16 f32. lanes 0-15 hold K=0,1; lanes 16-31 hold K=2,3 ----
  const float kx = KP[3 * p + 0];
  const float ky = KP[3 * p + 1];
  const float kz = KP[3 * p + 2];
  const float ksq = kx * kx + ky * ky + kz * kz;

  v2f bfrag;
  bfrag.x = hi ? (2.0f * s * kz) : (2.0f * s * kx);
  bfrag.y = hi ? (-s)            : (2.0f * s * ky);

  // ---- C fragment: -s*||kp_p||^2 broadcast down the M column ----
  const float cval = -s * ksq;
  v8f cfrag;
  #pragma unroll
  for (int i = 0; i < 8; ++i) cfrag[i] = cval;

  float acc = 0.0f;
  const long rowbase = (long)wave * kK;

  #pragma unroll
  for (int t = 0; t < 2; ++t) {
    const long row = rowbase + t * 16 + p;   // M = p (= lane % 16)
    const float* xr = X + row * 3;

    float ax, ay;
    if (!hi) { ax = xr[0]; ay = xr[1]; }     // K=0, K=1
    else     { ax = xr[2]; ay = 0.0f; }      // K=2, (K=3 filled below)

    // Build ||x_m||^2 in A[m][3] (lives in hi lanes, second VGPR).
    float partial = ax * ax + ay * ay;       // lo: x0^2+x1^2 ; hi: x2^2
    float other   = __shfl_xor(partial, 16, 32);
    if (hi) ay = other + ax * ax;            // full ||x_m||^2

    v2f afrag; afrag.x = ax; afrag.y = ay;

    v8f d = __builtin_amdgcn_wmma_f32_16x16x4_f32(
        false, afrag, false, bfrag, (short)0, cfrag, false, false);

    // exp of the 8 accumulator elements (M = r or r+8 for this lane, N = p)
    #pragma unroll
    for (int i = 0; i < 8; ++i) acc += __expf(d[i]);
  }

  // Sum the two half-tiles of M (lanes l and l+16 hold M=0..7 / M=8..15),
  // then mean over the K=32 neighbors (both tiles already in acc).
  acc += __shfl_xor(acc, 16, 32);
  if (!hi) out[(long)wave * kP + p] = acc * (1.0f / 32.0f);
}

extern "C" void kernel_launch(void* const* d_in, const int* in_sizes, int n_in,
                              void* d_out, int out_size, void* d_ws, size_t ws_size,
                              hipStream_t stream) {
  const float* X  = (const float*)d_in[0];   // neighborhoods (8,8192,32,3) f32
  const float* KP = (const float*)d_in[1];   // kernel_points (16,3) f32
  const float* sg = (const float*)d_in[2];   // sigma scalar
  float* out = (float*)d_out;                // (8,8192,16) f32

  const int blocks = kBN / kWavesPerBlk;     // 8192 blocks x 8 waves = 65536 waves
  gaussian_kernel_corr<<<blocks, kBlockDim, 0, stream>>>(X, KP, sg, out);
}